// QuantizedEmbedding_13460427506049
// MI455X (gfx1250) — compile-verified
//
#include <hip/hip_runtime.h>

// QuantizedEmbedding: out[t, d] = code[weight[x[t], d]] * absmax[x[t] >> 2]
//   x:      [16384] int32 token ids in [0, 128000)
//   weight: [128000, 1024] uint8 codes stored as int32
//   absmax: [32000] fp32 (one per 4096 flat elements -> one per 4 rows)
//   code:   [256] fp32 codebook
//   out:    [16384, 1024] fp32
//
// Memory-bound: ~64 MB random-row read + ~64 MB streaming write -> ~5.5 us at
// 23.3 TB/s. Strategy: B128 vector loads/stores with non-temporal hints (keep
// the 512 MB table stream out of L2's way), codebook in LDS (ds_load_b32),
// global_prefetch_b8 on the next token's row to hide random-gather latency.

typedef int   v4i __attribute__((ext_vector_type(4)));
typedef float v4f __attribute__((ext_vector_type(4)));

#define EMB_D      1024          // row length (elements)
#define EMB_V4     (EMB_D / 4)   // 256 v4i / v4f chunks per row
#define EMB_BLOCK  256           // threads per block == chunks per row

__global__ __launch_bounds__(EMB_BLOCK) void qembed_gather_dequant(
    const int*   __restrict__ x,
    const int*   __restrict__ weight,   // codes as int32
    const float* __restrict__ absmax,
    const float* __restrict__ code,
    float*       __restrict__ out,
    int ntok)
{
    __shared__ float lds_code[256];

    const int tid = threadIdx.x;
    // One codebook entry per thread (blockDim == 256) -> ds_store_b32
    lds_code[tid] = code[tid];
    __syncthreads();

    const v4i* __restrict__ w4 = (const v4i*)weight;

    for (int t = blockIdx.x; t < ntok; t += gridDim.x) {
        const int  idx = x[t];
        // Row of 1024 elems never straddles a 4096-elem quant block.
        const float a = absmax[idx >> 2];

        // 16 B of codes for this thread: global_load_b128 (non-temporal:
        // random rows over 512 MB, don't pollute L2 for the next gather).
        const long rowv4 = ((long)idx << 8);          // idx * 256 v4i chunks
        v4i w = __builtin_nontemporal_load(&w4[rowv4 + tid]);

        // Prefetch next token's row chunk (global_prefetch_b8); covers the
        // full 4 KB row across the 256 threads while we dequantize this one.
        const int tn = t + gridDim.x;
        if (tn < ntok) {
            const long nrow = ((long)x[tn] << 10);    // idx * 1024 ints
            __builtin_prefetch(weight + nrow + (tid << 2), 0, 0);
        }

        // 4x LDS codebook lookups (ds_load_b32) + scale.
        v4f r;
        r.x = lds_code[w.x & 255] * a;
        r.y = lds_code[w.y & 255] * a;
        r.z = lds_code[w.z & 255] * a;
        r.w = lds_code[w.w & 255] * a;

        // Streaming output, never re-read: global_store_b128 non-temporal.
        v4f* __restrict__ o4 = (v4f*)(out + ((long)t << 10));
        __builtin_nontemporal_store(r, &o4[tid]);
    }
}

extern "C" void kernel_launch(void* const* d_in, const int* in_sizes, int n_in,
                              void* d_out, int out_size, void* d_ws, size_t ws_size,
                              hipStream_t stream) {
    const int*   x      = (const int*)d_in[0];
    const int*   weight = (const int*)d_in[1];
    const float* absmax = (const float*)d_in[2];
    const float* code   = (const float*)d_in[3];
    float*       out    = (float*)d_out;

    const int ntok = in_sizes[0];                 // 4 * 4096 = 16384 tokens
    int grid = 2048;                              // 8 tokens per block
    if (grid > ntok) grid = ntok;

    qembed_gather_dequant<<<grid, EMB_BLOCK, 0, stream>>>(
        x, weight, absmax, code, out, ntok);
}